// StructuredFluxCorrectionLayer_81844896793248
// MI455X (gfx1250) — compile-verified
//
#include <hip/hip_runtime.h>
#include <hip/hip_bf16.h>
#include <math.h>

#ifndef __has_builtin
#define __has_builtin(x) 0
#endif

#if __has_builtin(__builtin_amdgcn_global_load_async_to_lds_b32)
#define USE_ASYNC_LDS 1
#else
#define USE_ASYNC_LDS 0
#endif
#if __has_builtin(__builtin_amdgcn_s_wait_asynccnt)
#define HAVE_WAIT_BUILTIN 1
#else
#define HAVE_WAIT_BUILTIN 0
#endif

#define Bn    8
#define Sn    512
#define ITERS 20
#define TS_I  8
#define TS_J  32

// match reference float semantics: python double constants used against f32 arrays
#define HF   ((float)(1.0/511.0))
#define H2F  ((float)(2.0*(1.0/511.0)))
#define HSQ  ((float)((1.0/511.0)*(1.0/511.0)))
#define EPSf 1e-6f

typedef __attribute__((address_space(1))) int glb_i32;
typedef __attribute__((address_space(3))) int lds_i32;

__device__ __forceinline__ int tid256() { return threadIdx.y * TS_J + threadIdx.x; }

__device__ __forceinline__ void async_ld(const float* g, float* l) {
#if USE_ASYNC_LDS
  __builtin_amdgcn_global_load_async_to_lds_b32((glb_i32*)g, (lds_i32*)l, 0, 0);
#else
  *l = *g;
#endif
}

__device__ __forceinline__ void async_wait() {
#if USE_ASYNC_LDS
#if HAVE_WAIT_BUILTIN
  __builtin_amdgcn_s_wait_asynccnt(0);
#else
  asm volatile("s_wait_asynccnt 0" ::: "memory");
#endif
#endif
}

__device__ __forceinline__ float block_sum(float v, float* red) {
  int t = tid256();
  red[t] = v;
  __syncthreads();
#pragma unroll
  for (int s = 128; s > 0; s >>= 1) {
    if (t < s) red[t] += red[t + s];
    __syncthreads();
  }
  float r = red[0];
  __syncthreads();
  return r;
}

// replicate np.linspace + np.argmin(first-min) boundary helpers in double
__device__ __forceinline__ void cell_bdry(int i, int j, float& rb, float& bnx, float& bny) {
  const double step = 1.0 / 511.0;
  double gx = (i == Sn - 1) ? 1.0 : (double)i * step;
  double gy = (j == Sn - 1) ? 1.0 : (double)j * step;
  double v0 = gx, v1 = 1.0 - gx, v2 = gy, v3 = 1.0 - gy;
  double d = fmin(fmin(v0, v1), fmin(v2, v3));
  int nearest = 0; double m = v0;
  if (v1 < m) { m = v1; nearest = 1; }
  if (v2 < m) { m = v2; nearest = 2; }
  if (v3 < m) { m = v3; nearest = 3; }
  float df = (float)d;
  const float dd = (float)(0.15 * 0.15);
  rb = expf(-(df * df) / dd);
  bnx = (nearest == 0) ? -1.f : ((nearest == 1) ? 1.f : 0.f);
  bny = (nearest == 2) ? -1.f : ((nearest == 3) ? 1.f : 0.f);
}

__global__ void k_zero(float* sc) {
  int t = blockIdx.x * blockDim.x + threadIdx.x;
  if (t < (2 * ITERS + 1) * Bn) sc[t] = 0.f;
}

__global__ __launch_bounds__(256) void k_prep(
    const float* __restrict__ coeff, const float* __restrict__ u,
    const float* __restrict__ beta, float* __restrict__ lap,
    float* __restrict__ cx, float* __restrict__ cy, float* __restrict__ rho,
    float* __restrict__ Px, float* __restrict__ Py) {
  const int b = blockIdx.z;
  const int i = blockIdx.y * TS_I + threadIdx.y;
  const int j = blockIdx.x * TS_J + threadIdx.x;
  const size_t idx = (size_t)b * Sn * Sn + (size_t)i * Sn + j;

  float lv = 0.f;
  if (i > 0 && i < Sn - 1 && j > 0 && j < Sn - 1) {
    float uc = u[idx];
    lv = ((u[idx + Sn] - uc) / HF - (uc - u[idx - Sn]) / HF) / HF +
         ((u[idx + 1] - uc) / HF - (uc - u[idx - 1]) / HF) / HF;
  }
  lap[idx] = lv;

  float cc = coeff[idx];
  if (i < Sn - 1) { float cn = coeff[idx + Sn]; cx[idx] = 2.f * cn * cc / (cn + cc + EPSf); }
  if (j < Sn - 1) { float cn = coeff[idx + 1];  cy[idx] = 2.f * cn * cc / (cn + cc + EPSf); }

  float glx, gly;
  {
    float Lc = logf(fmaxf(cc, EPSf));
    if (i == 0)           glx = (logf(fmaxf(coeff[idx + Sn], EPSf)) - Lc) / HF;
    else if (i == Sn - 1) glx = (Lc - logf(fmaxf(coeff[idx - Sn], EPSf))) / HF;
    else                  glx = (logf(fmaxf(coeff[idx + Sn], EPSf)) -
                                 logf(fmaxf(coeff[idx - Sn], EPSf))) / H2F;
    if (j == 0)           gly = (logf(fmaxf(coeff[idx + 1], EPSf)) - Lc) / HF;
    else if (j == Sn - 1) gly = (Lc - logf(fmaxf(coeff[idx - 1], EPSf))) / HF;
    else                  gly = (logf(fmaxf(coeff[idx + 1], EPSf)) -
                                 logf(fmaxf(coeff[idx - 1], EPSf))) / H2F;
  }
  float eta = sqrtf(glx * glx + gly * gly + EPSf);
  rho[idx] = 1.f / (1.f + expf(-10.0f * (eta - 0.5f)));
  float nhx = glx / eta, nhy = gly / eta;
  float bnv = beta[idx * 4 + 1], btv = beta[idx * 4 + 2];
  Px[idx] = bnv * nhx - btv * nhy;
  Py[idx] = bnv * nhy + btv * nhx;
}

__global__ __launch_bounds__(256) void k_flux(
    const float* __restrict__ u, const float* __restrict__ beta,
    const float* __restrict__ lap, const float* __restrict__ cx,
    const float* __restrict__ cy, const float* __restrict__ rho,
    const float* __restrict__ Px, const float* __restrict__ Py,
    float* __restrict__ tx, float* __restrict__ ty) {
  const int b = blockIdx.z;
  const int i = blockIdx.y * TS_I + threadIdx.y;
  const int j = blockIdx.x * TS_J + threadIdx.x;
  const size_t idx = (size_t)b * Sn * Sn + (size_t)i * Sn + j;

  float rb0, bnx0, bny0;
  cell_bdry(i, j, rb0, bnx0, bny0);

  if (i < Sn - 1) {  // x-face between cells (i,j) and (i+1,j)
    float v = 0.f;
    if (j >= 1 && j <= Sn - 2) {
      size_t q = idx + Sn;
      float rb1, bnx1, bny1; cell_bdry(i + 1, j, rb1, bnx1, bny1);
      float cxv = cx[idx];
      float gux = (u[q] - u[idx]) / HF;
      float lgx = (lap[q] - lap[idx]) / HF;
      float bulk = 0.5f * (beta[idx * 4 + 0] + beta[q * 4 + 0]) * HSQ * cxv * lgx;
      float intr = 0.5f * (rho[idx] + rho[q]) * (0.5f * (Px[idx] + Px[q])) * cxv * gux;
      float bdry = 0.5f * (rb0 + rb1) *
                   (0.5f * (beta[idx * 4 + 3] * bnx0 + beta[q * 4 + 3] * bnx1)) * cxv * gux;
      v = bulk + intr + bdry;
    }
    tx[idx] = v;
  }
  if (j < Sn - 1) {  // y-face between cells (i,j) and (i,j+1)
    float v = 0.f;
    if (i >= 1 && i <= Sn - 2) {
      size_t q = idx + 1;
      float rb1, bnx1, bny1; cell_bdry(i, j + 1, rb1, bnx1, bny1);
      float cyv = cy[idx];
      float guy = (u[q] - u[idx]) / HF;
      float lgy = (lap[q] - lap[idx]) / HF;
      float bulk = 0.5f * (beta[idx * 4 + 0] + beta[q * 4 + 0]) * HSQ * cyv * lgy;
      float intr = 0.5f * (rho[idx] + rho[q]) * (0.5f * (Py[idx] + Py[q])) * cyv * guy;
      float bdry = 0.5f * (rb0 + rb1) *
                   (0.5f * (beta[idx * 4 + 3] * bny0 + beta[q * 4 + 3] * bny1)) * cyv * guy;
      v = bulk + intr + bdry;
    }
    ty[idx] = v;
  }
}

__global__ __launch_bounds__(256) void k_rhs(
    const float* __restrict__ tx, const float* __restrict__ ty,
    float* __restrict__ r, float* __restrict__ p, float* __restrict__ x,
    float* __restrict__ rr0) {
  __shared__ float red[256];
  const int b = blockIdx.z;
  const int i = blockIdx.y * TS_I + threadIdx.y;
  const int j = blockIdx.x * TS_J + threadIdx.x;
  const size_t idx = (size_t)b * Sn * Sn + (size_t)i * Sn + j;
  float v = 0.f;
  if (i > 0 && i < Sn - 1 && j > 0 && j < Sn - 1)
    v = (tx[idx] - tx[idx - Sn]) / HF + (ty[idx] - ty[idx - 1]) / HF;
  r[idx] = v; p[idx] = v; x[idx] = 0.f;
  float s = block_sum(v * v, red);
  if (tid256() == 0) atomicAdd(&rr0[b], s);
}

// Fused: p_new = r + beta*p_old (LDS tiles, async-filled), ap = A(p_new),
// pAp partial reduce. Double-buffered p: reads pin, writes pout (no race).
__global__ __launch_bounds__(256) void k_applyf(
    float* __restrict__ ap, const float* __restrict__ pin, float* __restrict__ pout,
    const float* __restrict__ r, const float* __restrict__ cx,
    const float* __restrict__ cy, float* __restrict__ pap_k,
    const float* __restrict__ rr_k, const float* __restrict__ rr_km1, int first) {
  __shared__ float tp[TS_I + 2][TS_J + 2];
  __shared__ float tr[TS_I + 2][TS_J + 2];
  __shared__ float red[256];
  const int b = blockIdx.z;
  const int i0 = blockIdx.y * TS_I, j0 = blockIdx.x * TS_J;
  const size_t base = (size_t)b * Sn * Sn;
  const float* pb = pin + base;
  const float* rb = r + base;
  const int t = tid256();

  for (int q = t; q < (TS_I + 2) * (TS_J + 2); q += TS_I * TS_J) {
    int ti = q / (TS_J + 2);
    int tj = q - ti * (TS_J + 2);
    int gi = min(Sn - 1, max(0, i0 + ti - 1));
    int gj = min(Sn - 1, max(0, j0 + tj - 1));
    size_t off = (size_t)gi * Sn + gj;
    async_ld(pb + off, &tp[ti][tj]);
    if (!first) async_ld(rb + off, &tr[ti][tj]);
  }
  async_wait();
  __syncthreads();

  if (!first) {  // form p_new = r + beta*p_old in LDS (identical in all blocks)
    float beta = rr_k[b] / fmaxf(rr_km1[b], EPSf);
    for (int q = t; q < (TS_I + 2) * (TS_J + 2); q += TS_I * TS_J) {
      int ti = q / (TS_J + 2);
      int tj = q - ti * (TS_J + 2);
      tp[ti][tj] = tr[ti][tj] + beta * tp[ti][tj];
    }
    __syncthreads();
  }

  const int i = i0 + threadIdx.y, j = j0 + threadIdx.x;
  const size_t idx = base + (size_t)i * Sn + j;
  float pc = tp[threadIdx.y + 1][threadIdx.x + 1];
  float apv;
  if (i == 0 || i == Sn - 1 || j == 0 || j == Sn - 1) {
    apv = pc;  // Dirichlet rows: A(p) = p
  } else {
    float pim = tp[threadIdx.y][threadIdx.x + 1];
    float pip = tp[threadIdx.y + 2][threadIdx.x + 1];
    float pjm = tp[threadIdx.y + 1][threadIdx.x];
    float pjp = tp[threadIdx.y + 1][threadIdx.x + 2];
    float qxp = cx[idx] * ((pip - pc) / HF);
    float qxm = cx[idx - Sn] * ((pc - pim) / HF);
    float qyp = cy[idx] * ((pjp - pc) / HF);
    float qym = cy[idx - 1] * ((pc - pjm) / HF);
    apv = -((qxp - qxm) / HF + (qyp - qym) / HF);
  }
  pout[idx] = pc;
  ap[idx] = apv;
  float s = block_sum(pc * apv, red);
  if (t == 0) atomicAdd(&pap_k[b], s);
}

__global__ __launch_bounds__(256) void k_update(
    float* __restrict__ x, float* __restrict__ r, const float* __restrict__ p,
    const float* __restrict__ ap, const float* __restrict__ rr_k,
    float* __restrict__ rr_k1, const float* __restrict__ pap_k) {
  __shared__ float red[256];
  const int b = blockIdx.z;
  const int i = blockIdx.y * TS_I + threadIdx.y;
  const int j = blockIdx.x * TS_J + threadIdx.x;
  const size_t idx = (size_t)b * Sn * Sn + (size_t)i * Sn + j;
  float alpha = rr_k[b] / fmaxf(pap_k[b], EPSf);
  float xn = x[idx] + alpha * p[idx];
  if (i == 0 || i == Sn - 1 || j == 0 || j == Sn - 1) xn = 0.f;
  x[idx] = xn;
  float rn = r[idx] - alpha * ap[idx];
  r[idx] = rn;
  float s = block_sum(rn * rn, red);
  if (tid256() == 0) atomicAdd(&rr_k1[b], s);
}

__global__ __launch_bounds__(256) void k_final(float* __restrict__ x,
                                               const float* __restrict__ u) {
  const int b = blockIdx.z;
  const int i = blockIdx.y * TS_I + threadIdx.y;
  const int j = blockIdx.x * TS_J + threadIdx.x;
  const size_t idx = (size_t)b * Sn * Sn + (size_t)i * Sn + j;
  float v = 0.f;
  if (i > 0 && i < Sn - 1 && j > 0 && j < Sn - 1) v = u[idx] + x[idx];
  x[idx] = v;
}

extern "C" void kernel_launch(void* const* d_in, const int* in_sizes, int n_in,
                              void* d_out, int out_size, void* d_ws, size_t ws_size,
                              hipStream_t stream) {
  (void)in_sizes; (void)n_in; (void)out_size; (void)ws_size;
  const float* coeff = (const float*)d_in[0];
  const float* u     = (const float*)d_in[1];
  const float* beta  = (const float*)d_in[2];
  float* x = (float*)d_out;

  const size_t N = (size_t)Bn * Sn * Sn;
  float* w   = (float*)d_ws;
  float* lap = w;           // reused as r after flux/rhs
  float* cx  = w + 1 * N;
  float* cy  = w + 2 * N;
  float* rho = w + 3 * N;   // reused as p (buffer A)
  float* Px  = w + 4 * N;   // reused as ap
  float* Py  = w + 5 * N;   // reused as p (buffer B, ping-pong)
  float* tx  = w + 6 * N;
  float* ty  = w + 7 * N;
  float* pap = w + 8 * N;          // ITERS*Bn
  float* rr  = pap + ITERS * Bn;   // (ITERS+1)*Bn
  float* r = lap; float* ap = Px;

  dim3 blk(TS_J, TS_I, 1);
  dim3 grd(Sn / TS_J, Sn / TS_I, Bn);

  k_zero<<<1, 512, 0, stream>>>(pap);
  k_prep<<<grd, blk, 0, stream>>>(coeff, u, beta, lap, cx, cy, rho, Px, Py);
  k_flux<<<grd, blk, 0, stream>>>(u, beta, lap, cx, cy, rho, Px, Py, tx, ty);
  k_rhs <<<grd, blk, 0, stream>>>(tx, ty, r, rho /*p0*/, x, rr);

  float* pcur = rho;  // holds p_0 = rhs
  float* palt = Py;
  for (int k = 0; k < ITERS; ++k) {
    k_applyf<<<grd, blk, 0, stream>>>(ap, pcur, palt, r, cx, cy,
                                      pap + (size_t)k * Bn, rr + (size_t)k * Bn,
                                      (k > 0) ? rr + (size_t)(k - 1) * Bn : rr,
                                      (k == 0) ? 1 : 0);
    float* tmp = pcur; pcur = palt; palt = tmp;
    k_update<<<grd, blk, 0, stream>>>(x, r, pcur, ap, rr + (size_t)k * Bn,
                                      rr + (size_t)(k + 1) * Bn, pap + (size_t)k * Bn);
  }
  k_final<<<grd, blk, 0, stream>>>(x, u);
}